// FlatDilation1D_6270652252754
// MI455X (gfx1250) — compile-verified
//
#include <hip/hip_runtime.h>
#include <stdint.h>

#define K_SIZE 41
#define BLOCK 256

typedef __attribute__((ext_vector_type(4))) unsigned int v4u;
typedef __attribute__((ext_vector_type(8))) int          v8i;
typedef __attribute__((ext_vector_type(4))) int          v4i;

// out[i] = max_k( win[(i+k) % 41] + h[k] ),  win = in[(n-41)/2 - 2 ...],
// h[k] = -((k-20)/scale)^16.  Output is periodic with period 41.
__global__ __launch_bounds__(BLOCK) void FlatDilation1D_kernel(
    const float* __restrict__ in, const float* __restrict__ scale_p,
    float* __restrict__ out, int n) {
  __shared__ float win[K_SIZE];
  __shared__ float hk[K_SIZE];
  __shared__ float res[K_SIZE];

  const int t = threadIdx.x;
  const int start = (n - K_SIZE) / 2 - 2;  // 4073 for n=8191

#if defined(__gfx1250__) && __has_builtin(__builtin_amdgcn_tensor_load_to_lds) && \
    __has_builtin(__builtin_amdgcn_s_wait_tensorcnt)
  // Tensor Data Mover: DMA the 41-dword window from global to LDS.
  if (t == 0) {
    unsigned long long ga = (unsigned long long)(uintptr_t)(const void*)(in + start);
    unsigned int ga_lo = (unsigned int)(ga & 0xFFFFFFFFull);
    unsigned int ga_hi = (unsigned int)(ga >> 32);
    // Generic pointer to LDS = SHARED_BASE[63:32] | byte-offset[31:0].
    unsigned int lds_off = (unsigned int)(uintptr_t)(void*)&win[0];

    v4u g0;
    g0.x = 1u;                                   // count=1, is_restore=0, gather off
    g0.y = lds_off;                              // lds_addr (bytes)
    g0.z = ga_lo;                                // global_addr[31:0]
    g0.w = (ga_hi & 0x01FFFFFFu) | 0x80000000u;  // global_addr[56:32] | type=2

    v8i g1;
    g1[0] = (int)(2u << 16);                     // workgroup_mask=0, data_size=4B
    g1[1] = (int)((unsigned)K_SIZE << 16);       // tensor_dim0 = 41 (low 16 bits)
    g1[2] = (int)(1u << 16);                     // tensor_dim1 = 1
    g1[3] = (int)((unsigned)K_SIZE << 16);       // tile_dim0 = 41
    g1[4] = 0;                                   // tile_dim1=0 (1-D), tile_dim2=0
    g1[5] = K_SIZE;                              // tensor_dim0_stride = 41
    g1[6] = 0;
    g1[7] = 0;

    v4i g2 = {0, 0, 0, 0};
    v4i g3 = {0, 0, 0, 0};
    v8i g4 = {0, 0, 0, 0, 0, 0, 0, 0};           // unused trailing group (zeros)
    __builtin_amdgcn_tensor_load_to_lds(g0, g1, g2, g3, g4, /*cpol=*/0);
    __builtin_amdgcn_s_wait_tensorcnt(0);
  }
#else
  if (t < K_SIZE) win[t] = in[start + t];
#endif

  // h[k] = -((k-20)/s)^16, computed by 4 squarings (matches lax.integer_pow).
  const float s = scale_p[0];
  if (t < K_SIZE) {
    float r  = (float)(t - K_SIZE / 2) / s;
    float r2 = r * r;
    float r4 = r2 * r2;
    float r8 = r4 * r4;
    hk[t] = -(r8 * r8);
  }
  __syncthreads();

  // 41 unique results (output has period 41).
  if (t < K_SIZE) {
    float acc = -__builtin_huge_valf();
#pragma unroll
    for (int k = 0; k < K_SIZE; ++k) {
      int j = t + k;
      if (j >= K_SIZE) j -= K_SIZE;
      acc = fmaxf(acc, win[j] + hk[k]);
    }
    res[t] = acc;
  }
  __syncthreads();

  const int i = blockIdx.x * BLOCK + t;
  if (i < n) out[i] = res[i % K_SIZE];
}

extern "C" void kernel_launch(void* const* d_in, const int* in_sizes, int n_in,
                              void* d_out, int out_size, void* d_ws, size_t ws_size,
                              hipStream_t stream) {
  (void)n_in; (void)d_ws; (void)ws_size; (void)out_size;
  const float* in      = (const float*)d_in[0];
  const float* scale_p = (const float*)d_in[1];
  float*       out     = (float*)d_out;
  const int n = in_sizes[0];
  const int blocks = (n + BLOCK - 1) / BLOCK;
  hipLaunchKernelGGL(FlatDilation1D_kernel, dim3(blocks), dim3(BLOCK), 0, stream,
                     in, scale_p, out, n);
}